// RNNunit_35605278883922
// MI455X (gfx1250) — compile-verified
//
#include <hip/hip_runtime.h>
#include <hip/hip_bf16.h>
#include <stdint.h>

// ---------------------------------------------------------------------------
// RNN scan:  Xp = X @ Wx  (f32 WMMA GEMM, register-blocked 64x32/wave,
//                          written in-place into d_out)
//            h_t = relu(Xp[t] + h_{t-1} @ Wh + bh)
//            (persistent scan, Wh pinned in 16 WGPs' LDS, device barrier/step)
// Sizes: T=16384, M1=M2=1024.
// ---------------------------------------------------------------------------

#define T_STEPS 16384
#define MDIM    1024

typedef float v2f __attribute__((ext_vector_type(2)));
typedef float v8f __attribute__((ext_vector_type(8)));

// ------------------------- Phase 1: Xp = X @ Wx (WMMA) ----------------------
// Each wave computes a 64x32 strip: 4 M-tiles x 2 N-tiles of 16x16.
// 8 independent accumulator chains; A frags shared across N, B frags across M.
#define MT 4   // 16-row tiles per wave (M)
#define NT2 2  // 16-col tiles per wave (N)

__global__ void __launch_bounds__(256)
xp_gemm_wmma_f32(const float* __restrict__ X,
                 const float* __restrict__ Wx,
                 float* __restrict__ Xp) {
  const int wave  = threadIdx.x >> 5;
  const int lane  = threadIdx.x & 31;
  const int strip = blockIdx.x * 8 + wave;          // 8192 strips total
  const int NS    = MDIM / (16 * NT2);              // 32 strips along N
  const int sm    = strip / NS;                     // 0..255
  const int sn    = strip - sm * NS;                // 0..31
  const int m0    = sm * (16 * MT);                 // 64-row strip
  const int n0    = sn * (16 * NT2);                // 32-col strip

  const int lhalf = lane >> 4;                      // 0: lanes 0-15, 1: 16-31
  const int l     = lane & 15;

  const float* __restrict__ Arow[MT];
#pragma unroll
  for (int mi = 0; mi < MT; ++mi)
    Arow[mi] = X + (size_t)(m0 + 16 * mi + l) * MDIM;

  v8f acc[MT][NT2] = {};

  for (int k = 0; k < MDIM; k += 4) {
    // A 16x4 f32 fragment per M-tile: VGPR v holds K = v + 2*lhalf
    const int ka = k + 2 * lhalf;
    v2f a[MT];
#pragma unroll
    for (int mi = 0; mi < MT; ++mi) {
      a[mi].x = Arow[mi][ka + 0];
      a[mi].y = Arow[mi][ka + 1];
    }
    // B 4x16 f32 fragment per N-tile: VGPR v holds row K = v + 2*lhalf
    v2f b[NT2];
#pragma unroll
    for (int ni = 0; ni < NT2; ++ni) {
      b[ni].x = Wx[(size_t)(ka + 0) * MDIM + n0 + 16 * ni + l];
      b[ni].y = Wx[(size_t)(ka + 1) * MDIM + n0 + 16 * ni + l];
    }
#pragma unroll
    for (int mi = 0; mi < MT; ++mi)
#pragma unroll
      for (int ni = 0; ni < NT2; ++ni)
        acc[mi][ni] = __builtin_amdgcn_wmma_f32_16x16x4_f32(
            false, a[mi], false, b[ni],
            (short)0, acc[mi][ni], false, false);
  }

  // C/D layout: VGPR r -> row +r + 8*lhalf, col +l
#pragma unroll
  for (int mi = 0; mi < MT; ++mi)
#pragma unroll
    for (int ni = 0; ni < NT2; ++ni)
#pragma unroll
      for (int r = 0; r < 8; ++r)
        Xp[(size_t)(m0 + 16 * mi + r + 8 * lhalf) * MDIM + n0 + 16 * ni + l] =
            acc[mi][ni][r];
}

// ------------------------- Phase 2: sequential scan -------------------------
#define NBLK      16
#define COLS      64                         // columns per block
#define KPAD      (MDIM + 1)                 // bank-conflict-free stride
#define LDS_WH    (COLS * KPAD)              // 65600 floats
#define LDS_H     MDIM                       // 1024 floats
#define LDS_PART  256                        // per-thread partials
#define SCAN_LDS_BYTES ((LDS_WH + LDS_H + LDS_PART) * sizeof(float))  // 267,520 B

struct GBar { unsigned cnt; unsigned gen; };

__global__ void scan_init(const float* __restrict__ h0,
                          float* __restrict__ hbuf0,
                          GBar* bar) {
  if (threadIdx.x == 0) { bar->cnt = 0u; bar->gen = 0u; }
  for (int i = threadIdx.x; i < MDIM; i += blockDim.x) hbuf0[i] = h0[i];
}

__global__ void __launch_bounds__(256)
rnn_scan_lds(const float* __restrict__ Wh,
             const float* __restrict__ bh,
             float* __restrict__ out,       // holds Xp on entry; h_t on exit
             float* __restrict__ hbuf0,
             float* __restrict__ hbuf1,
             GBar* __restrict__ bar) {
  extern __shared__ float smem[];
  float* wh_lds   = smem;                    // [COLS][KPAD]
  float* h_lds    = smem + LDS_WH;           // [MDIM]
  float* part_lds = smem + LDS_WH + LDS_H;   // [256]

  const int tid  = threadIdx.x;
  const int jj   = tid & (COLS - 1);         // column within slice
  const int q    = tid >> 6;                 // k-quarter (0..3)
  const int col0 = blockIdx.x * COLS;

  // --- Stage this block's Wh column slice into LDS (once) ---
  for (int idx = tid; idx < COLS * MDIM; idx += 256) {
    const int k = idx >> 6;                  // 0..1023
    const int c = idx & (COLS - 1);
    wh_lds[c * KPAD + k] = Wh[(size_t)k * MDIM + col0 + c];
  }
  const float bias = (tid < COLS) ? bh[col0 + tid] : 0.0f;
  __syncthreads();

  const float* __restrict__ wcol = &wh_lds[jj * KPAD];
  const int kbase = q * 256;

  for (int t = 0; t < T_STEPS; ++t) {
    const float* __restrict__ hcur = (t & 1) ? hbuf1 : hbuf0;
    float* __restrict__ hnxt       = (t & 1) ? hbuf0 : hbuf1;

    // broadcast current hidden state into LDS (coalesced float4)
    {
      const float4* __restrict__ src = (const float4*)hcur;
      ((float4*)h_lds)[tid] = src[tid];
    }
    __syncthreads();

    // partial dot: quarter q of column col0+jj (LDS-resident weights)
    float acc = 0.0f;
#pragma unroll 8
    for (int kk = 0; kk < 256; ++kk) {
      acc = fmaf(h_lds[kbase + kk], wcol[kbase + kk], acc);
    }
    part_lds[tid] = acc;
    __syncthreads();

    if (tid < COLS) {
      const int j = col0 + tid;
      const float dot = part_lds[tid] + part_lds[tid + 64] +
                        part_lds[tid + 128] + part_lds[tid + 192];
      const float xp = out[(size_t)t * MDIM + j];   // Xp[t][j]
      const float hv = fmaxf(xp + dot + bias, 0.0f);
      out[(size_t)t * MDIM + j] = hv;               // overwrite with h_t
      hnxt[j] = hv;
      // warm next step's Xp row while the barrier settles (global_prefetch_b8)
      if (t + 1 < T_STEPS)
        __builtin_prefetch(&out[(size_t)(t + 1) * MDIM + j], 0, 1);
    }
    __syncthreads();
    __threadfence();

    // device-wide generation barrier (16 arrivals)
    if (tid == 0) {
      const unsigned prev = __hip_atomic_fetch_add(&bar->cnt, 1u,
                              __ATOMIC_ACQ_REL, __HIP_MEMORY_SCOPE_AGENT);
      if (prev == NBLK - 1) {
        __hip_atomic_store(&bar->cnt, 0u,
                           __ATOMIC_RELAXED, __HIP_MEMORY_SCOPE_AGENT);
        __hip_atomic_fetch_add(&bar->gen, 1u,
                               __ATOMIC_RELEASE, __HIP_MEMORY_SCOPE_AGENT);
      } else {
        while (__hip_atomic_load(&bar->gen, __ATOMIC_ACQUIRE,
                                 __HIP_MEMORY_SCOPE_AGENT) < (unsigned)(t + 1)) {
          __builtin_amdgcn_s_sleep(1);
        }
      }
    }
    __syncthreads();
  }
}

// ------------------------------- launcher -----------------------------------
extern "C" void kernel_launch(void* const* d_in, const int* in_sizes, int n_in,
                              void* d_out, int out_size, void* d_ws, size_t ws_size,
                              hipStream_t stream) {
  const float* X  = (const float*)d_in[0];   // [16384,1024]
  const float* Wx = (const float*)d_in[1];   // [1024,1024]
  const float* Wh = (const float*)d_in[2];   // [1024,1024]
  const float* bh = (const float*)d_in[3];   // [1024]
  const float* h0 = (const float*)d_in[4];   // [1,1024]
  float* out = (float*)d_out;                // [16384,1024]

  uint8_t* ws   = (uint8_t*)d_ws;
  GBar*  bar    = (GBar*)ws;
  float* hbuf0  = (float*)(ws + 256);
  float* hbuf1  = (float*)(ws + 256 + MDIM * sizeof(float));

  (void)in_sizes; (void)n_in; (void)out_size; (void)ws_size;

  hipFuncSetAttribute((const void*)rnn_scan_lds,
                      hipFuncAttributeMaxDynamicSharedMemorySize,
                      (int)SCAN_LDS_BYTES);

  // Phase 1: Xp = X @ Wx -> d_out
  // 8192 strips of 64x32, 8 waves/block -> 1024 blocks.
  const int nstrips = (T_STEPS / (16 * MT)) * (MDIM / (16 * NT2));
  xp_gemm_wmma_f32<<<nstrips / 8, 256, 0, stream>>>(X, Wx, out);

  // Phase 2: reset barrier + h, then persistent scan
  scan_init<<<1, 256, 0, stream>>>(h0, hbuf0, bar);
  rnn_scan_lds<<<NBLK, 256, SCAN_LDS_BYTES, stream>>>(Wh, bh, out, hbuf0, hbuf1, bar);
}